// DeepHitsurvLoss_68728066671351
// MI455X (gfx1250) — compile-verified
//
#include <hip/hip_runtime.h>
#include <math.h>

// DeepHit survival loss, reformulated:
//   nll: per-row log-sum-exp/cumsum stats (memory-bound, ~8.4MB read)
//   rank-loss: exact algebraic collapse of the 8192x8192 pairwise term into
//              Q = M^T (256x8192, {0,1}) x P (8192x256, fp32)  -- a real GEMM
//              executed with V_WMMA_F32_16X16X4_F32 (fp32-exact), split-K
//              across up to 32 chunks for occupancy (up to 8192 wave32s),
//              then a deterministic fixed-order partial reduction.
// All reductions are fixed-order (deterministic), no float atomics.

#define TT 256            // time bins (reference T)
#define EPSV 1e-7f

typedef __attribute__((ext_vector_type(2))) float v2f;
typedef __attribute__((ext_vector_type(8))) float v8f;

// ---------------------------------------------------------------------------
// Kernel 1: one block per subject row j (256 threads = 8 waves).
//  phi = [hazards_row, 0]; gamma = max(phi); csum = cumsum(exp(phi-gamma));
//  sum_ = csum[256]; cdf[t] = csum[t]/sum_;  P[j,t] = exp(2*cdf[j,t]).
//  Also emits per-row NLL contribution, ae[j] = ev_j*exp(-2*cdf[j,lab_j]),
//  and thr[j] = dur_j + (ev_j==0)  so that  m_j[d] = (d < thr_j).
// ---------------------------------------------------------------------------
__global__ void row_stats_kernel(const float* __restrict__ hz,
                                 const int* __restrict__ dur,
                                 const int* __restrict__ ev,
                                 const int* __restrict__ lab,
                                 float* __restrict__ P,
                                 int* __restrict__ thr,
                                 float* __restrict__ nll_buf,
                                 float* __restrict__ ae_buf) {
  const int j = blockIdx.x;
  const int t = threadIdx.x;
  __shared__ float sm[TT];
  __shared__ float sc[TT];

  const float h = hz[(size_t)j * TT + t];

  // block max (pad column contributes 0.0f)
  sm[t] = h;
  __syncthreads();
  for (int off = TT / 2; off > 0; off >>= 1) {
    if (t < off) sm[t] = fmaxf(sm[t], sm[t + off]);
    __syncthreads();
  }
  const float gamma = fmaxf(sm[0], 0.0f);
  __syncthreads();

  // inclusive scan of exp(h - gamma) (Hillis-Steele)
  float e = expf(h - gamma);
  sc[t] = e;
  __syncthreads();
  for (int off = 1; off < TT; off <<= 1) {
    float v = (t >= off) ? sc[t - off] : 0.0f;
    __syncthreads();
    sc[t] += v;
    __syncthreads();
  }
  const float sum_ = sc[TT - 1] + expf(-gamma);  // + pad term exp(0-gamma)

  const float cdf = sc[t] / sum_;
  P[(size_t)j * TT + t] = expf(2.0f * cdf);      // 1/SIGMA = 2

  if (t == 0) {
    const int   L      = lab[j];                 // L in [0, TT)
    const float evf    = (ev[j] != 0) ? 1.0f : 0.0f;
    const float cum_at = sc[L];
    const float phi_at = hz[(size_t)j * TT + L];
    const float part1  = (phi_at - gamma) * evf;
    const float part2  = -logf(sum_ + EPSV);     // sum_ > 0 always
    const float part3  = logf(fmaxf(sum_ - cum_at, 0.0f) + EPSV) * (1.0f - evf);
    nll_buf[j] = -(part1 + part2 + part3);
    const float a = cum_at / sum_;
    ae_buf[j] = evf * expf(-2.0f * a);
    thr[j] = dur[j] + ((ev[j] == 0) ? 1 : 0);
  }
}

// ---------------------------------------------------------------------------
// Kernel 2: split-K partial GEMM  Qpart[z] = M^T[:, jB:jE] x P[jB:jE, :]
//  Grid (TT/16, TT/16, S), one wave32 per block; each wave owns a 16x16 tile
//  of one K-chunk, stepping K by 8 with two independent accumulator chains.
//  A fragment (16x4 f32): lane<16 -> M=lane,   K = {0,1}
//                         lane>=16 -> M=lane-16, K = {2,3}
//  B fragment (4x16 f32): lanes 0-15 hold K rows {0,1}, lanes 16-31 rows {2,3}.
// ---------------------------------------------------------------------------
__global__ void rank_gemm_kernel(const float* __restrict__ P,
                                 const int* __restrict__ thr,
                                 float* __restrict__ Qpart,
                                 int kChunk) {
  const int lane  = threadIdx.x;      // 0..31
  const int half  = lane >> 4;        // 0 or 1
  const int col   = lane & 15;
  const int tBase = blockIdx.x * 16;  // N (label/time column)
  const int dBase = blockIdx.y * 16;  // M (duration row)
  const int d     = dBase + col;      // this lane's A row index
  const int jB    = blockIdx.z * kChunk;
  const int jE    = jB + kChunk;

  v8f c0 = {0.f, 0.f, 0.f, 0.f, 0.f, 0.f, 0.f, 0.f};
  v8f c1 = {0.f, 0.f, 0.f, 0.f, 0.f, 0.f, 0.f, 0.f};

  for (int j0 = jB; j0 < jE; j0 += 8) {
    // ---- chain 0: K block [j0, j0+4) ----
    {
      const int jj  = j0 + (half << 1);
      const int th0 = thr[jj];
      const int th1 = thr[jj + 1];
      v2f a, b;
      a.x = (d < th0) ? 1.0f : 0.0f;
      a.y = (d < th1) ? 1.0f : 0.0f;
      b.x = P[(size_t)jj * TT + tBase + col];
      b.y = P[(size_t)(jj + 1) * TT + tBase + col];
      c0 = __builtin_amdgcn_wmma_f32_16x16x4_f32(
          false, a, false, b, (short)0, c0, false, false);
    }
    // ---- chain 1: K block [j0+4, j0+8) ----
    {
      const int jj  = j0 + 4 + (half << 1);
      const int th0 = thr[jj];
      const int th1 = thr[jj + 1];
      v2f a, b;
      a.x = (d < th0) ? 1.0f : 0.0f;
      a.y = (d < th1) ? 1.0f : 0.0f;
      b.x = P[(size_t)jj * TT + tBase + col];
      b.y = P[(size_t)(jj + 1) * TT + tBase + col];
      c1 = __builtin_amdgcn_wmma_f32_16x16x4_f32(
          false, a, false, b, (short)0, c1, false, false);
    }
  }

  // C/D layout: element k of accumulator -> row M = k + 8*half, col N = col
  float* Qz = Qpart + (size_t)blockIdx.z * TT * TT;
#pragma unroll
  for (int k = 0; k < 8; ++k) {
    Qz[(size_t)(dBase + k + 8 * half) * TT + (tBase + col)] = c0[k] + c1[k];
  }
}

// ---------------------------------------------------------------------------
// Kernel 2b: deterministic reduction of split-K partials: Q = sum_z Qpart[z]
// ---------------------------------------------------------------------------
__global__ void qreduce_kernel(const float* __restrict__ Qpart,
                               float* __restrict__ Q,
                               int S) {
  const int idx = blockIdx.x * blockDim.x + threadIdx.x;  // 0 .. TT*TT-1
  float acc = 0.0f;
  for (int s = 0; s < S; ++s) {
    acc += Qpart[(size_t)s * TT * TT + idx];
  }
  Q[idx] = acc;
}

// ---------------------------------------------------------------------------
// Kernel 3: deterministic final reduction (single block, fixed order).
//  loss = 0.5 * mean(nll) + 0.5 * (1/n^2) * sum_i ae[i] * Q[dur_i, lab_i]
// ---------------------------------------------------------------------------
__global__ void finalize_kernel(const float* __restrict__ nll_buf,
                                const float* __restrict__ ae_buf,
                                const float* __restrict__ Q,
                                const int* __restrict__ dur,
                                const int* __restrict__ lab,
                                float* __restrict__ out,
                                int n) {
  __shared__ float s1[256];
  __shared__ float s2[256];
  const int t = threadIdx.x;
  float aN = 0.0f, aR = 0.0f;
  for (int i = t; i < n; i += 256) {
    aN += nll_buf[i];
    aR += ae_buf[i] * Q[(size_t)dur[i] * TT + lab[i]];
  }
  s1[t] = aN;
  s2[t] = aR;
  __syncthreads();
  for (int off = 128; off > 0; off >>= 1) {
    if (t < off) {
      s1[t] += s1[t + off];
      s2[t] += s2[t + off];
    }
    __syncthreads();
  }
  if (t == 0) {
    const float nll = s1[0] / (float)n;
    const float rl  = s2[0] / ((float)n * (float)n);
    out[0] = 0.5f * nll + 0.5f * rl;
  }
}

// ---------------------------------------------------------------------------
extern "C" void kernel_launch(void* const* d_in, const int* in_sizes, int n_in,
                              void* d_out, int out_size, void* d_ws, size_t ws_size,
                              hipStream_t stream) {
  const float* hz  = (const float*)d_in[0];  // (n, 256) fp32
  const int*   dur = (const int*)d_in[1];    // (n,) int32 in [0,256)
  const int*   ev  = (const int*)d_in[2];    // (n,) int32 in {0,1}
  const int*   lab = (const int*)d_in[3];    // (n,) int32 in [0,256)
  float* out = (float*)d_out;

  const int n = in_sizes[1];                 // 8192

  // Workspace layout (fp32 unless noted):
  //   P       : n*TT floats     (8.0 MB)
  //   thr     : n ints          (32 KB)
  //   nll_buf : n floats        (32 KB)
  //   ae_buf  : n floats        (32 KB)
  //   Q       : TT*TT floats    (256 KB)
  //   Qpart   : S*TT*TT floats  (S chosen to fit ws_size; up to 8 MB)
  char* ws = (char*)d_ws;
  const size_t offThr  = (size_t)n * TT * 4;
  const size_t offNll  = offThr + (size_t)n * 4;
  const size_t offAe   = offNll + (size_t)n * 4;
  const size_t offQ    = offAe  + (size_t)n * 4;
  const size_t offQp   = offQ   + (size_t)TT * TT * 4;

  float* P       = (float*)ws;
  int*   thr     = (int*)(ws + offThr);
  float* nll_buf = (float*)(ws + offNll);
  float* ae_buf  = (float*)(ws + offAe);
  float* Q       = (float*)(ws + offQ);
  float* Qpart   = (float*)(ws + offQp);

  // Split-K factor: largest power-of-two <= 32 whose partial buffer fits.
  // Deterministic: depends only on ws_size (constant across calls).
  int S = 1;
  for (int cand = 32; cand >= 1; cand >>= 1) {
    if (offQp + (size_t)cand * TT * TT * 4 <= ws_size && (n % (cand * 8)) == 0) {
      S = cand;
      break;
    }
  }
  const int kChunk = n / S;

  row_stats_kernel<<<n, TT, 0, stream>>>(hz, dur, ev, lab, P, thr, nll_buf, ae_buf);

  dim3 gemm_grid(TT / 16, TT / 16, S);
  rank_gemm_kernel<<<gemm_grid, 32, 0, stream>>>(P, thr, Qpart, kChunk);

  qreduce_kernel<<<(TT * TT) / 256, 256, 0, stream>>>(Qpart, Q, S);

  finalize_kernel<<<1, 256, 0, stream>>>(nll_buf, ae_buf, Q, dur, lab, out, n);
}